// Eckstein2024Model_76922864271487
// MI455X (gfx1250) — compile-verified
//
#include <hip/hip_runtime.h>
#include <cstdint>

// Problem constants (T, B, NA) from the reference.
#define TT   1024
#define BB   4096
#define NACT 4
#define CH   16          // timesteps per scan chunk
#define NCH  (TT / CH)   // 64 chunks
#define SCH  4           // timesteps staged per async group
#define NST  (CH / SCH)  // stage groups per chunk
#define TPB  256

// gfx1250 async global->LDS path, guarded so the file compiles even if the
// builtin names differ on this toolchain (fallback: plain b128 loads).
#if defined(__HIP_DEVICE_COMPILE__) && \
    __has_builtin(__builtin_amdgcn_global_load_async_to_lds_b128) && \
    __has_builtin(__builtin_amdgcn_s_wait_asynccnt)
#define USE_ASYNC 1
#else
#define USE_ASYNC 0
#endif

__device__ __forceinline__ float sigmoidf_(float x) {
    return 1.0f / (1.0f + __expf(-x));
}

#if USE_ASYNC
// Builtin signature (from compiler diagnostic): parameters are pointers to
// int __vector_size__(16) in global / LDS address spaces.
typedef int v4i_ __attribute__((__vector_size__(16)));
typedef __attribute__((address_space(1))) v4i_ gv4i;
typedef __attribute__((address_space(3))) v4i_ lv4i;

// Stage SCH consecutive timesteps (32B each = act4 + rew4) for this lane's
// batch element into LDS buffer `buf` via async DMA. 2 x b128 per step.
__device__ __forceinline__ void stage_async(const float* __restrict__ inp,
                                            float4 (&tile)[2][SCH][TPB][2],
                                            int buf, int tstart, int b, int tid) {
#pragma unroll
    for (int s = 0; s < SCH; ++s) {
        const float* g = inp + ((size_t)(tstart + s) * BB + (size_t)b) * 8;
        gv4i* gp = (gv4i*)g;
        lv4i* lp = (lv4i*)&tile[buf][s][tid][0];
        __builtin_amdgcn_global_load_async_to_lds_b128(gp, lp, 0, 0);
        __builtin_amdgcn_global_load_async_to_lds_b128(gp, lp, 16, 0);
    }
}
#endif

// Per-step affine composition: q' = m*q + c  (per action channel).
__device__ __forceinline__ void affine_step(float4 a, float4 r,
                                            float al, float one_m_al, float af,
                                            float4& M, float4& C) {
    float rv   = fmaf(a.x, r.x, fmaf(a.y, r.y, fmaf(a.z, r.z, a.w * r.w)));
    float alrv = al * rv;
    float m0 = (a.x > 0.5f) ? one_m_al : af;  float c0 = (a.x > 0.5f) ? alrv : 0.0f;
    float m1 = (a.y > 0.5f) ? one_m_al : af;  float c1 = (a.y > 0.5f) ? alrv : 0.0f;
    float m2 = (a.z > 0.5f) ? one_m_al : af;  float c2 = (a.z > 0.5f) ? alrv : 0.0f;
    float m3 = (a.w > 0.5f) ? one_m_al : af;  float c3 = (a.w > 0.5f) ? alrv : 0.0f;
    M.x *= m0;  C.x = fmaf(m0, C.x, c0);
    M.y *= m1;  C.y = fmaf(m1, C.y, c1);
    M.z *= m2;  C.z = fmaf(m2, C.z, c2);
    M.w *= m3;  C.w = fmaf(m3, C.w, c3);
}

// Full reference step: emit logits, update q, return current choice.
__device__ __forceinline__ int emit_step(float4 a, float4 r, float beta, float pp,
                                         float al, float afm1, int prev,
                                         float4& q, float* outp) {
    float4 l;
    l.x = beta * (q.x + ((prev == 0) ? pp : 0.0f));
    l.y = beta * (q.y + ((prev == 1) ? pp : 0.0f));
    l.z = beta * (q.z + ((prev == 2) ? pp : 0.0f));
    l.w = beta * (q.w + ((prev == 3) ? pp : 0.0f));
    *(float4*)outp = l;
    float rv = fmaf(a.x, r.x, fmaf(a.y, r.y, fmaf(a.z, r.z, a.w * r.w)));
    q.x = q.x + al * (rv - q.x) * a.x + afm1 * q.x * (1.0f - a.x);
    q.y = q.y + al * (rv - q.y) * a.y + afm1 * q.y * (1.0f - a.y);
    q.z = q.z + al * (rv - q.z) * a.z + afm1 * q.z * (1.0f - a.z);
    q.w = q.w + al * (rv - q.w) * a.w + afm1 * q.w * (1.0f - a.w);
    int cur = 0;
    if (a.y > 0.5f) cur = 1;
    if (a.z > 0.5f) cur = 2;
    if (a.w > 0.5f) cur = 3;
    return cur;
}

__device__ __forceinline__ int argmax4(float4 a) {
    int cur = 0;
    if (a.y > 0.5f) cur = 1;
    if (a.z > 0.5f) cur = 2;
    if (a.w > 0.5f) cur = 3;
    return cur;
}

// K1: per (chunk, batch) thread, compose the chunk's affine transform (M, C).
__global__ __launch_bounds__(TPB) void k1_chunk_affine(const float* __restrict__ inp,
                                                       const float* __restrict__ alp,
                                                       const float* __restrict__ afp,
                                                       float* __restrict__ mc) {
    const int tid = threadIdx.x;
    const int b   = blockIdx.x * TPB + tid;
    const int c   = blockIdx.y;
    const int t0  = c * CH;
    const float al  = sigmoidf_(alp[0]);
    const float af  = sigmoidf_(afp[0]);
    const float oma = 1.0f - al;
    float4 M = make_float4(1.0f, 1.0f, 1.0f, 1.0f);
    float4 C = make_float4(0.0f, 0.0f, 0.0f, 0.0f);

#if USE_ASYNC
    __shared__ float4 tile[2][SCH][TPB][2];   // 64 KB, double-buffered
    stage_async(inp, tile, 0, t0, b, tid);
#pragma unroll
    for (int j = 0; j < NST; ++j) {
        if (j + 1 < NST) {
            stage_async(inp, tile, (j + 1) & 1, t0 + (j + 1) * SCH, b, tid);
            __builtin_amdgcn_s_wait_asynccnt(2 * SCH);  // current buffer landed
        } else {
            __builtin_amdgcn_s_wait_asynccnt(0);
        }
#pragma unroll
        for (int s = 0; s < SCH; ++s) {
            float4 a = tile[j & 1][s][tid][0];
            float4 r = tile[j & 1][s][tid][1];
            affine_step(a, r, al, oma, af, M, C);
        }
    }
#else
#pragma unroll 4
    for (int s = 0; s < CH; ++s) {
        const float4* g = (const float4*)(inp + ((size_t)(t0 + s) * BB + (size_t)b) * 8);
        affine_step(g[0], g[1], al, oma, af, M, C);
    }
#endif

    float4* o = (float4*)(mc + ((size_t)c * BB + (size_t)b) * 8);
    o[0] = M;
    o[1] = C;
}

// K2: sequential scan over chunks (only 64 iterations), emits q at every
// chunk boundary plus the final q (a reference output).
__global__ __launch_bounds__(TPB) void k2_scan(const float* __restrict__ mc,
                                               float* __restrict__ qstart,
                                               float* __restrict__ qfinal) {
    const int b = blockIdx.x * TPB + threadIdx.x;
    float4 q = make_float4(0.0f, 0.0f, 0.0f, 0.0f);
#pragma unroll 4
    for (int c = 0; c < NCH; ++c) {
        const float4* p4 = (const float4*)(mc + ((size_t)c * BB + (size_t)b) * 8);
        float4 M = p4[0];
        float4 C = p4[1];
        *(float4*)(qstart + ((size_t)c * BB + (size_t)b) * 4) = q;
        q.x = fmaf(M.x, q.x, C.x);
        q.y = fmaf(M.y, q.y, C.y);
        q.z = fmaf(M.z, q.z, C.z);
        q.w = fmaf(M.w, q.w, C.w);
    }
    *(float4*)(qfinal + (size_t)b * 4) = q;
}

// K3: per (chunk, batch) thread, replay the chunk from its boundary q and
// emit logits; prev_choice is data-local (argmax of previous step's action).
__global__ __launch_bounds__(TPB) void k3_emit(const float* __restrict__ inp,
                                               const float* __restrict__ qstart,
                                               const float* __restrict__ bp,
                                               const float* __restrict__ alp,
                                               const float* __restrict__ afp,
                                               const float* __restrict__ ppp,
                                               float* __restrict__ out) {
    const int tid = threadIdx.x;
    const int b   = blockIdx.x * TPB + tid;
    const int c   = blockIdx.y;
    const int t0  = c * CH;
    const float beta = bp[0];
    const float pp   = ppp[0];
    const float al   = sigmoidf_(alp[0]);
    const float afm1 = sigmoidf_(afp[0]) - 1.0f;

    float4 q = *(const float4*)(qstart + ((size_t)c * BB + (size_t)b) * 4);
    int prev = -1;
    if (c > 0) {
        const float4* g = (const float4*)(inp + ((size_t)(t0 - 1) * BB + (size_t)b) * 8);
        prev = argmax4(g[0]);
    }

#if USE_ASYNC
    __shared__ float4 tile[2][SCH][TPB][2];
    stage_async(inp, tile, 0, t0, b, tid);
#pragma unroll
    for (int j = 0; j < NST; ++j) {
        if (j + 1 < NST) {
            stage_async(inp, tile, (j + 1) & 1, t0 + (j + 1) * SCH, b, tid);
            __builtin_amdgcn_s_wait_asynccnt(2 * SCH);
        } else {
            __builtin_amdgcn_s_wait_asynccnt(0);
        }
#pragma unroll
        for (int s = 0; s < SCH; ++s) {
            const int t = t0 + j * SCH + s;
            float4 a = tile[j & 1][s][tid][0];
            float4 r = tile[j & 1][s][tid][1];
            prev = emit_step(a, r, beta, pp, al, afm1, prev, q,
                             out + ((size_t)t * BB + (size_t)b) * 4);
        }
    }
#else
#pragma unroll 4
    for (int s = 0; s < CH; ++s) {
        const int t = t0 + s;
        const float4* g = (const float4*)(inp + ((size_t)t * BB + (size_t)b) * 8);
        prev = emit_step(g[0], g[1], beta, pp, al, afm1, prev, q,
                         out + ((size_t)t * BB + (size_t)b) * 4);
    }
#endif

    if (c == NCH - 1) {
        out[(size_t)TT * BB * NACT + (size_t)BB * NACT + b] = (float)prev;
    }
}

extern "C" void kernel_launch(void* const* d_in, const int* in_sizes, int n_in,
                              void* d_out, int out_size, void* d_ws, size_t ws_size,
                              hipStream_t stream) {
    const float* inp  = (const float*)d_in[0];  // (T, B, 2*NA) f32
    const float* beta = (const float*)d_in[1];
    const float* all  = (const float*)d_in[2];
    const float* afl  = (const float*)d_in[3];
    const float* p    = (const float*)d_in[4];
    float* out = (float*)d_out;

    const size_t mc_floats = (size_t)NCH * BB * 8;  // 8 MB
    const size_t qs_floats = (size_t)NCH * BB * 4;  // 4 MB

    // Scratch plan (deterministic, fixed by ws_size):
    //  - If d_ws is big enough, keep (M,C) and boundary-q both in d_ws.
    //  - Otherwise stash (M,C) in the logits region of d_out (it is fully
    //    overwritten by k3 afterwards) and keep only boundary-q in d_ws.
    const bool big_ws = ws_size >= (mc_floats + qs_floats) * sizeof(float);
    float* mc     = big_ws ? (float*)d_ws : out;
    float* qstart = big_ws ? ((float*)d_ws + mc_floats) : (float*)d_ws;
    float* qfinal = out + (size_t)TT * BB * NACT;

    dim3 gridMap(BB / TPB, NCH);
    k1_chunk_affine<<<gridMap, TPB, 0, stream>>>(inp, all, afl, mc);
    k2_scan<<<BB / TPB, TPB, 0, stream>>>(mc, qstart, qfinal);
    k3_emit<<<gridMap, TPB, 0, stream>>>(inp, qstart, beta, all, afl, p, out);
}